// GATAgg_89343909691966
// MI455X (gfx1250) — compile-verified
//
#include <hip/hip_runtime.h>
#include <hip/hip_bf16.h>
#include <math.h>

#define NNODES 50000
#define MNB 16
#define DD 128

typedef __attribute__((ext_vector_type(2))) float v2f;
typedef __attribute__((ext_vector_type(8))) float v8f;

__device__ __forceinline__ float readlane_f32(float v, int lane) {
  return __int_as_float(__builtin_amdgcn_readlane(__float_as_int(v), lane));
}

// ---------------------------------------------------------------------------
// Kernel 0 (tiny): c_self[k] = sum_j a[j]     * W[j,k]
//                  c_nb[k]   = sum_j a[128+j] * W[j,k]
// Collapses h = x@W.T followed by dots with a_self/a_nb into two 128-vectors.
// ---------------------------------------------------------------------------
__global__ void gat_proj_vecs(const float* __restrict__ W,
                              const float* __restrict__ a,
                              float* __restrict__ c /* 256 floats */) {
  const int k = threadIdx.x;           // 0..127, coalesced over W rows
  float cs = 0.f, cn = 0.f;
  for (int j = 0; j < DD; ++j) {
    const float w = W[j * DD + k];
    cs = fmaf(a[j], w, cs);
    cn = fmaf(a[DD + j], w, cn);
  }
  c[k]      = cs;
  c[DD + k] = cn;
}

// ---------------------------------------------------------------------------
// Kernel 1: S = X(16-node tile x 128) @ C(128x16, cols 0/1 = c_self/c_nb)
// via chained V_WMMA_F32_16X16X4_F32. One wave per tile, 32 K-steps.
// A 16x4 f32 fragment: lane&15 = row M, lane>>4 selects K={0,1}/{2,3}.
// B 4x16 fragment:     lane&15 = col N, VGPR r holds K=r (lanes 0-15),
//                      K=r+2 (lanes 16-31) -> same (k0 + 2*kh) offset as A.
// B columns >= 2 are zeroed branchlessly (clamped load * 0/1 mask) so the
// loop body has no EXEC manipulation and loads pipeline freely.
// D: column n lives in lane n (rows 0-7, VGPR r=row) and lane n+16 (rows 8-15).
// ---------------------------------------------------------------------------
__global__ void __launch_bounds__(256)
gat_scores_wmma(const float* __restrict__ x,
                const float* __restrict__ c,
                float* __restrict__ s,
                float* __restrict__ t) {
  const int lane = threadIdx.x & 31;
  const int wave = threadIdx.x >> 5;
  const int tile = blockIdx.x * 8 + wave;      // 16 nodes per tile
  const int ntiles = NNODES / 16;              // 3125 exactly
  if (tile >= ntiles) return;                  // wave-uniform: EXEC stays full

  const int n16 = lane & 15;                   // A-row / B-col / D-col index
  const int kh2 = (lane >> 4) * 2;             // K half select (0 or 2)
  const long rowBase = (long)(tile * 16 + n16) * DD;

  // Branchless B: lanes with column >= 2 read column 0 (valid memory) and
  // multiply by 0; lanes 0/1 read c_self/c_nb and multiply by 1.
  const float bm = (n16 < 2) ? 1.f : 0.f;
  const float* bsrc = c + ((n16 < 2) ? n16 : 0) * DD + kh2;
  const float* asrc = x + rowBase + kh2;

  v8f acc = {};
  #pragma unroll 8
  for (int k0 = 0; k0 < DD; k0 += 4) {
    const v2f av   = *(const v2f*)(asrc + k0);
    const v2f braw = *(const v2f*)(bsrc + k0);
    v2f bv;
    bv.x = braw.x * bm;
    bv.y = braw.y * bm;
    acc = __builtin_amdgcn_wmma_f32_16x16x4_f32(
        /*neg_a=*/false, av, /*neg_b=*/false, bv,
        /*c_mod=*/(short)0, acc, /*reuse_a=*/false, /*reuse_b=*/false);
  }

  // Column 0 -> s_self, column 1 -> t. Lanes {0,16} and {1,17} hold them.
  if (n16 < 2) {
    float* dst = (n16 == 0) ? s : t;
    const int nodeBase = tile * 16 + (kh2 >> 1) * 8;  // upper lanes: rows 8..15
    #pragma unroll
    for (int r = 0; r < 8; ++r) dst[nodeBase + r] = acc[r];
  }
}

// ---------------------------------------------------------------------------
// Kernel 2: one wave per node. Lanes 0-15 each own one neighbor:
// gather t[j], e = leaky_relu(s[i]+t[j]), shuffle-softmax over width 16.
// Broadcast (j_m, alpha_m) via v_readlane -> scalar row base, so the 16
// row gathers are SGPR-base + lane-offset b128 loads (fully coalesced),
// and alpha_m is a scalar FMA operand. ELU + float4 store on the way out.
// ---------------------------------------------------------------------------
__global__ void __launch_bounds__(256)
gat_agg(const float* __restrict__ x,
        const int* __restrict__ xnb,
        const float* __restrict__ s,
        const float* __restrict__ t,
        float* __restrict__ out) {
  const int lane = threadIdx.x & 31;
  const int wave = threadIdx.x >> 5;
  const int i = blockIdx.x * 8 + wave;
  if (i >= NNODES) return;                     // wave-uniform

  int   j = 0;
  float e = 0.f;
  if (lane < MNB) {
    j = xnb[i * MNB + lane] - 1;
    const float v = s[i] + t[j];
    e = (v >= 0.f) ? v : 0.01f * v;            // leaky_relu, slope 0.01
  }

  // softmax over lanes 0..15 (lanes 16-31 run harmless identity reductions)
  float mx = e;
  #pragma unroll
  for (int off = 8; off >= 1; off >>= 1)
    mx = fmaxf(mx, __shfl_xor(mx, off, 16));
  const float ex = __expf(e - mx);
  float sum = ex;
  #pragma unroll
  for (int off = 8; off >= 1; off >>= 1)
    sum += __shfl_xor(sum, off, 16);
  const float alpha = ex / sum;

  float4 acc = make_float4(0.f, 0.f, 0.f, 0.f);
  #pragma unroll
  for (int m = 0; m < MNB; ++m) {
    const int   jm = __builtin_amdgcn_readlane(j, m);   // SGPR broadcast
    const float am = readlane_f32(alpha, m);            // SGPR broadcast
    const float4* row = (const float4*)(x + (long)jm * DD);
    const float4 z = row[lane];                         // coalesced 512B row
    acc.x = fmaf(am, z.x, acc.x);
    acc.y = fmaf(am, z.y, acc.y);
    acc.z = fmaf(am, z.z, acc.z);
    acc.w = fmaf(am, z.w, acc.w);
  }

  float4 r;
  r.x = (acc.x > 0.f) ? acc.x : expm1f(acc.x); // elu, alpha=1
  r.y = (acc.y > 0.f) ? acc.y : expm1f(acc.y);
  r.z = (acc.z > 0.f) ? acc.z : expm1f(acc.z);
  r.w = (acc.w > 0.f) ? acc.w : expm1f(acc.w);
  ((float4*)out)[(long)i * (DD / 4) + lane] = r;
}

// ---------------------------------------------------------------------------
extern "C" void kernel_launch(void* const* d_in, const int* in_sizes, int n_in,
                              void* d_out, int out_size, void* d_ws, size_t ws_size,
                              hipStream_t stream) {
  const float* x   = (const float*)d_in[0];    // (N,128) f32
  const int*   xnb = (const int*)  d_in[1];    // (N,16)  int
  const float* W   = (const float*)d_in[2];    // (128,128) f32
  const float* a   = (const float*)d_in[3];    // (1,256) f32
  float* out = (float*)d_out;

  // workspace: c (256) | s (N) | t (N)   -> ~401 KB
  float* c = (float*)d_ws;
  float* s = c + 256;
  float* t = s + NNODES;

  gat_proj_vecs<<<1, DD, 0, stream>>>(W, a, c);

  const int ntiles = NNODES / 16;                       // 3125
  const int blocks1 = (ntiles + 7) / 8;                 // 8 waves/block
  gat_scores_wmma<<<blocks1, 256, 0, stream>>>(x, c, s, t);

  const int blocks2 = (NNODES + 7) / 8;                 // 6250
  gat_agg<<<blocks2, 256, 0, stream>>>(x, xnb, s, t, out);
}